// DisRNN_14233521619245
// MI455X (gfx1250) — compile-verified
//
#include <hip/hip_runtime.h>
#include <math.h>

// DisRNN forward: B=131072, Z=16, O=8, D=24, H=32, all fp32.
// One 512-thread block (16 wave32) per 16-row batch tile (x TPB tiles/block).
// Wave z runs latent-z's update MLP via V_WMMA_F32_16X16X4_F32; wave 0 then
// runs the global bottleneck + choice MLP for the tile.
// The dominant stream (eps_update, 201MB) is staged into LDS with
// GLOBAL_LOAD_ASYNC_TO_LDS_B128; z_tilde is written back with
// GLOBAL_STORE_ASYNC_FROM_LDS_B128.

#define B_TOT 131072
#define Zl 16
#define Ob 8
#define Dd 24          // Z + O
#define Hh 32
#define MT 16          // batch rows per tile (WMMA M)
#define TPB 4          // tiles per block
#define NWAVE 16
#define NTHREADS (NWAVE * 32)

#define EPS_ROW 384            // Dd*Zl floats per row
#define EPS_STRIDE 388         // padded LDS row stride (16B-aligned, stride%64==4 banks)
#define X_STRIDE 28            // 24 + 4 pad (16B-aligned)

typedef __attribute__((ext_vector_type(2))) float v2f;
typedef __attribute__((ext_vector_type(8))) float v8f;

__device__ __forceinline__ v8f wmma4(v2f a, v2f b, v8f c) {
  // D = A(16x4) * B(4x16) + C(16x16), fp32
  return __builtin_amdgcn_wmma_f32_16x16x4_f32(
      /*neg_a=*/false, a, /*neg_b=*/false, b,
      /*c_mod=*/(short)0, c, /*reuse_a=*/false, /*reuse_b=*/false);
}

__device__ __forceinline__ unsigned lds_off_of(const void* p) {
  // generic LDS pointer = {aperture_hi32, lds_offset32}; low 32 bits are the
  // byte offset used by the async engine's VDST/VSRC operand.
  return (unsigned)(unsigned long long)p;
}

__device__ __forceinline__ void async_ld_b128(unsigned lds_off, const void* gaddr) {
#if defined(__gfx1250__)
  asm volatile("global_load_async_to_lds_b128 %0, %1, off"
               :: "v"(lds_off), "v"(gaddr) : "memory");
#endif
}

__device__ __forceinline__ void async_st_b128(void* gaddr, unsigned lds_off) {
#if defined(__gfx1250__)
  asm volatile("global_store_async_from_lds_b128 %0, %1, off"
               :: "v"(gaddr), "v"(lds_off) : "memory");
#endif
}

__device__ __forceinline__ void wait_async0() {
#if defined(__gfx1250__)
  asm volatile("s_wait_asynccnt 0" ::: "memory");
#endif
}

__device__ __forceinline__ void wait_ds0() {
#if defined(__gfx1250__)
  asm volatile("s_wait_dscnt 0" ::: "memory");
#endif
}

__global__ __launch_bounds__(NTHREADS) void disrnn_fwd(
    const float* __restrict__ latents, const float* __restrict__ obs,
    const float* __restrict__ mul, const float* __restrict__ up_lv,
    const float* __restrict__ glob_lv,
    const float* __restrict__ uW1, const float* __restrict__ ub1,
    const float* __restrict__ uW2, const float* __restrict__ ub2,
    const float* __restrict__ uW3, const float* __restrict__ ub3,
    const float* __restrict__ cW1, const float* __restrict__ cb1,
    const float* __restrict__ cW2, const float* __restrict__ cb2,
    const float* __restrict__ cW3, const float* __restrict__ cb3,
    const float* __restrict__ eps_update, const float* __restrict__ eps_global,
    float* __restrict__ out_y, float* __restrict__ out_zt,
    float* __restrict__ out_kld)
{
  __shared__ float lds_h[NWAVE][MT][Hh + 1];              // D->A transpose staging
  __shared__ float lds_nl[MT][Zl + 1];                    // new_latents[row][z]
  __shared__ float lds_ukld[Zl][MT];                      // per-(z,row) update KLD
  __shared__ __align__(16) float eps_lds[MT][EPS_STRIDE]; // eps_update tile [row][d*16+z]
  __shared__ __align__(16) float x_lds[MT][X_STRIDE];     // concat(latents,obs) tile
  __shared__ __align__(16) float zt_lds[MT][Zl];          // z_tilde tile (store staging)

  const int tid  = threadIdx.x;
  const int wave = tid >> 5;       // == z for the update MLPs
  const int lane = tid & 31;
  const int n    = lane & 15;      // A: row index; B/D: column index
  const int half = lane >> 4;      // K-group selector within a fragment
  const int z    = wave;

  const unsigned eps_base = lds_off_of(&eps_lds[0][0]);
  const unsigned x_base   = lds_off_of(&x_lds[0][0]);
  const unsigned zt_base  = lds_off_of(&zt_lds[0][0]);

  for (int tile = 0; tile < TPB; ++tile) {
    const int base = (blockIdx.x * TPB + tile) * MT;

    // ---- async stage: eps tile (16x384 f32, contiguous) + x tile into LDS --
#pragma unroll
    for (int j = 0; j < 3; ++j) {                 // 1536 B128 chunks / 512 thr
      const int c   = tid + j * NTHREADS;
      const int row = c / 96;                     // 96 x 16B chunks per row
      const int q16 = c % 96;
      async_ld_b128(eps_base + (unsigned)(row * EPS_STRIDE + q16 * 4) * 4u,
                    eps_update + ((size_t)(base + row) * EPS_ROW + q16 * 4));
    }
    if (tid < 64) {                               // latents: 4 chunks/row
      const int row = tid >> 2, q = tid & 3;
      async_ld_b128(x_base + (unsigned)(row * X_STRIDE + q * 4) * 4u,
                    latents + (size_t)(base + row) * Zl + q * 4);
    } else if (tid < 96) {                        // obs: 2 chunks/row
      const int c = tid - 64, row = c >> 1, q = c & 1;
      async_ld_b128(x_base + (unsigned)(row * X_STRIDE + Zl + q * 4) * 4u,
                    obs + (size_t)(base + row) * Ob + q * 4);
    }
    wait_async0();
    __syncthreads();

    // ================= phase 1: per-latent bottleneck + update MLP =========
    // A fragments for layer1: x_tilde (16 rows x 24 d), 6 K-steps of 4.
    v2f a1[6];
    float kld_part = 0.f;
#pragma unroll
    for (int k = 0; k < 6; ++k) {
#pragma unroll
      for (int v = 0; v < 2; ++v) {
        const int d = 4 * k + v + 2 * half;
        const float xv   = x_lds[n][d];
        const float m    = mul[d * Zl + z];
        const float lv   = up_lv[d * Zl + z];
        const float mean = xv * m;
        kld_part += -lv - 1.0f + mean * mean + __expf(lv);
        const float e = eps_lds[n][d * Zl + z];
        a1[k][v] = e * __expf(0.5f * lv) + mean;  // x_tilde
      }
    }
    // lanes L and L^16 cover complementary d's of the same row
    kld_part += __shfl_xor(kld_part, 16, 32);
    if (lane < 16) lds_ukld[z][lane] = 0.5f * kld_part;

    // layer1: h1 = relu(x_tilde @ uW1[z]^T + ub1[z])  (N = 32 -> 2 tiles)
    v8f acc0 = {}; v8f acc1 = {};
#pragma unroll
    for (int k = 0; k < 6; ++k) {
      v2f b0, b1;
#pragma unroll
      for (int v = 0; v < 2; ++v) {
        const int d = 4 * k + v + 2 * half;
        b0[v] = uW1[z * Hh * Dd + (n     ) * Dd + d];
        b1[v] = uW1[z * Hh * Dd + (n + 16) * Dd + d];
      }
      acc0 = wmma4(a1[k], b0, acc0);
      acc1 = wmma4(a1[k], b1, acc1);
    }
    {
      const float bia0 = ub1[z * Hh + n];
      const float bia1 = ub1[z * Hh + n + 16];
#pragma unroll
      for (int v = 0; v < 8; ++v) {
        lds_h[z][v + 8 * half][n]      = fmaxf(acc0[v] + bia0, 0.f);
        lds_h[z][v + 8 * half][n + 16] = fmaxf(acc1[v] + bia1, 0.f);
      }
    }
    __syncthreads();

    // layer2: h2 = relu(h1 @ uW2[z]^T + ub2[z])
    v8f acc2_0 = {}; v8f acc2_1 = {};
#pragma unroll
    for (int k = 0; k < 8; ++k) {
      v2f a, b0, b1;
#pragma unroll
      for (int v = 0; v < 2; ++v) {
        const int h = 4 * k + v + 2 * half;
        a[v]  = lds_h[z][n][h];
        b0[v] = uW2[z * Hh * Hh + (n     ) * Hh + h];
        b1[v] = uW2[z * Hh * Hh + (n + 16) * Hh + h];
      }
      acc2_0 = wmma4(a, b0, acc2_0);
      acc2_1 = wmma4(a, b1, acc2_1);
    }
    __syncthreads();
    {
      const float bia0 = ub2[z * Hh + n];
      const float bia1 = ub2[z * Hh + n + 16];
#pragma unroll
      for (int v = 0; v < 8; ++v) {
        lds_h[z][v + 8 * half][n]      = fmaxf(acc2_0[v] + bia0, 0.f);
        lds_h[z][v + 8 * half][n + 16] = fmaxf(acc2_1[v] + bia1, 0.f);
      }
    }
    __syncthreads();

    // layer3: out[16x2] = h2 @ uW3[z]^T + ub3[z]  (N padded to 16 with zeros)
    v8f acc3 = {};
#pragma unroll
    for (int k = 0; k < 8; ++k) {
      v2f a, b;
#pragma unroll
      for (int v = 0; v < 2; ++v) {
        const int h = 4 * k + v + 2 * half;
        a[v] = lds_h[z][n][h];
        b[v] = (n < 2) ? uW3[z * 2 * Hh + n * Hh + h] : 0.f;
      }
      acc3 = wmma4(a, b, acc3);
    }
    {
      const float bia3 = (n < 2) ? ub3[z * 2 + n] : 0.f;
#pragma unroll
      for (int v = 0; v < 8; ++v) {
        const float valv = acc3[v] + bia3;
        // broadcast columns 0 (u) and 1 (pre-sigmoid w) of this lane-half
        const float uv = __shfl(valv, 16 * half + 0, 32);
        const float wv = __shfl(valv, 16 * half + 1, 32);
        if (n == 0) {
          const int row = v + 8 * half;
          const float w   = 1.0f / (1.0f + __expf(-wv));
          const float lat = x_lds[row][z];
          lds_nl[row][z] = (1.0f - w) * lat + uv * w;
        }
      }
    }
    __syncthreads();

    // ================ phase 2: global bottleneck + choice MLP (wave 0) =====
    v2f azt[4];
    if (wave == 0) {
      float gk_part = 0.f;
#pragma unroll
      for (int k = 0; k < 4; ++k) {
#pragma unroll
        for (int v = 0; v < 2; ++v) {
          const int zz  = 4 * k + v + 2 * half;
          const int row = n;
          const float glv = glob_lv[zz];
          const float nl  = lds_nl[row][zz];
          gk_part += -glv - 1.0f + nl * nl + __expf(glv);
          const float zt =
              eps_global[(size_t)(base + row) * Zl + zz] * __expf(0.5f * glv) + nl;
          azt[k][v] = zt;                                  // choice A fragment
          zt_lds[row][zz] = zt;                            // stage for async store
        }
      }
      gk_part += __shfl_xor(gk_part, 16, 32);
      if (lane < 16) {
        float s = 0.5f * gk_part;
#pragma unroll
        for (int zz = 0; zz < Zl; ++zz) s += lds_ukld[zz][lane];
        out_kld[base + lane] = s;
      }
      // drain this wave's DS stores, then write z_tilde tile coalesced
      wait_ds0();
#pragma unroll
      for (int j = 0; j < 2; ++j) {               // 64 B128 chunks / 32 lanes
        const int c = lane + 32 * j;
        const int row = c >> 2, q = c & 3;
        async_st_b128(out_zt + (size_t)(base + row) * Zl + q * 4,
                      zt_base + (unsigned)(row * Zl + q * 4) * 4u);
      }
      // choice layer1: z_tilde[16x16] @ cW1^T[16x32]
      v8f c0 = {}; v8f c1 = {};
#pragma unroll
      for (int k = 0; k < 4; ++k) {
        v2f b0, b1;
#pragma unroll
        for (int v = 0; v < 2; ++v) {
          const int zz = 4 * k + v + 2 * half;
          b0[v] = cW1[(n     ) * Zl + zz];
          b1[v] = cW1[(n + 16) * Zl + zz];
        }
        c0 = wmma4(azt[k], b0, c0);
        c1 = wmma4(azt[k], b1, c1);
      }
      const float bia0 = cb1[n];
      const float bia1 = cb1[n + 16];
#pragma unroll
      for (int v = 0; v < 8; ++v) {
        lds_h[0][v + 8 * half][n]      = fmaxf(c0[v] + bia0, 0.f);
        lds_h[0][v + 8 * half][n + 16] = fmaxf(c1[v] + bia1, 0.f);
      }
    }
    __syncthreads();
    if (wave == 0) {
      // choice layer2
      v8f c0 = {}; v8f c1 = {};
#pragma unroll
      for (int k = 0; k < 8; ++k) {
        v2f a, b0, b1;
#pragma unroll
        for (int v = 0; v < 2; ++v) {
          const int h = 4 * k + v + 2 * half;
          a[v]  = lds_h[0][n][h];
          b0[v] = cW2[(n     ) * Hh + h];
          b1[v] = cW2[(n + 16) * Hh + h];
        }
        c0 = wmma4(a, b0, c0);
        c1 = wmma4(a, b1, c1);
      }
      const float bia0 = cb2[n];
      const float bia1 = cb2[n + 16];
#pragma unroll
      for (int v = 0; v < 8; ++v) {
        lds_h[1][v + 8 * half][n]      = fmaxf(c0[v] + bia0, 0.f);
        lds_h[1][v + 8 * half][n + 16] = fmaxf(c1[v] + bia1, 0.f);
      }
    }
    __syncthreads();
    if (wave == 0) {
      // choice layer3: 32 -> 2 (N padded to 16)
      v8f c = {};
#pragma unroll
      for (int k = 0; k < 8; ++k) {
        v2f a, b;
#pragma unroll
        for (int v = 0; v < 2; ++v) {
          const int h = 4 * k + v + 2 * half;
          a[v] = lds_h[1][n][h];
          b[v] = (n < 2) ? cW3[n * Hh + h] : 0.f;
        }
        c = wmma4(a, b, c);
      }
      if (n < 2) {
        const float bia = cb3[n];
#pragma unroll
        for (int v = 0; v < 8; ++v) {
          const int row = v + 8 * half;
          out_y[(size_t)(base + row) * 2 + n] = c[v] + bia;
        }
      }
    }
    __syncthreads();
  }
}

extern "C" void kernel_launch(void* const* d_in, const int* in_sizes, int n_in,
                              void* d_out, int out_size, void* d_ws, size_t ws_size,
                              hipStream_t stream) {
  (void)in_sizes; (void)n_in; (void)out_size; (void)d_ws; (void)ws_size;
  const float* latents    = (const float*)d_in[0];
  const float* obs        = (const float*)d_in[1];
  const float* mul        = (const float*)d_in[2];
  const float* up_lv      = (const float*)d_in[3];
  const float* glob_lv    = (const float*)d_in[4];
  const float* uW1        = (const float*)d_in[5];
  const float* ub1        = (const float*)d_in[6];
  const float* uW2        = (const float*)d_in[7];
  const float* ub2        = (const float*)d_in[8];
  const float* uW3        = (const float*)d_in[9];
  const float* ub3        = (const float*)d_in[10];
  const float* cW1        = (const float*)d_in[11];
  const float* cb1        = (const float*)d_in[12];
  const float* cW2        = (const float*)d_in[13];
  const float* cb2        = (const float*)d_in[14];
  const float* cW3        = (const float*)d_in[15];
  const float* cb3        = (const float*)d_in[16];
  const float* eps_update = (const float*)d_in[17];
  const float* eps_global = (const float*)d_in[18];

  float* out     = (float*)d_out;
  float* out_y   = out;                        // [B,2]
  float* out_zt  = out + (size_t)B_TOT * 2;    // [B,16]
  float* out_kld = out + (size_t)B_TOT * 18;   // [B]

  dim3 grid(B_TOT / (MT * TPB));               // 2048 blocks
  disrnn_fwd<<<grid, NTHREADS, 0, stream>>>(
      latents, obs, mul, up_lv, glob_lv,
      uW1, ub1, uW2, ub2, uW3, ub3,
      cW1, cb1, cW2, cb2, cW3, cb3,
      eps_update, eps_global,
      out_y, out_zt, out_kld);
}